// EdgeNetwork_61057255080034
// MI455X (gfx1250) — compile-verified
//
#include <hip/hip_runtime.h>

typedef __attribute__((ext_vector_type(2))) float v2f;
typedef __attribute__((ext_vector_type(8))) float v8f;

#define N_ATOMS 8192
#define N_EDGES 65536
#define N_PAIR  16
#define NH      64
#define TILES   2                       // 16-edge tiles per wave
#define EDGES_PER_BLOCK (TILES * 16)    // 32 edges per block (4 waves cover h=0..63)

// Build Wt[p][k][h] = W[p, h*64+k] for p<16; p==16 holds transposed bias b[h*64+k].
__global__ void prep_wt(const float* __restrict__ W, const float* __restrict__ b,
                        float* __restrict__ Wt) {
    int idx = blockIdx.x * blockDim.x + threadIdx.x;
    if (idx >= 17 * NH * NH) return;
    int p = idx / (NH * NH);
    int r = idx % (NH * NH);
    int k = r / NH;
    int h = r % NH;
    Wt[idx] = (p < N_PAIR) ? W[p * (NH * NH) + h * NH + k] : b[h * NH + k];
}

__global__ void zero_out(float* __restrict__ out, int n) {
    int idx = blockIdx.x * blockDim.x + threadIdx.x;
    if (idx < n) out[idx] = 0.0f;
}

__launch_bounds__(128)
__global__ void edge_net(const float* __restrict__ pf,
                         const float* __restrict__ atomf,
                         const int*   __restrict__ a2p,
                         const float* __restrict__ Wt,
                         float*       __restrict__ out) {
    // +4 pad (68 floats/row) -> bank-conflict-free strided ds_load_b64 A-fragments
    __shared__ float srcLds[EDGES_PER_BLOCK][NH + 4];
    __shared__ float pfLds[EDGES_PER_BLOCK][N_PAIR + 1];
    __shared__ int   a0Lds[EDGES_PER_BLOCK];

    const int tid  = threadIdx.x;
    const int lane = tid & 31;
    const int wave = tid >> 5;
    const int eb   = blockIdx.x * EDGES_PER_BLOCK;

    // ---- stage gathered src rows (4 threads per edge, float4 chunks) ----
    {
        int el = tid >> 2;      // 0..31 local edge
        int q  = tid & 3;       // quarter of the 64-float row
        int e  = eb + el;
        int a1 = a2p[2 * e + 1];
        const float4* srow = (const float4*)(atomf + (size_t)a1 * NH);
#pragma unroll
        for (int j = 0; j < 4; ++j) {
            float4 v = srow[q * 4 + j];
            float* d = &srcLds[el][q * 16 + j * 4];
            d[0] = v.x; d[1] = v.y; d[2] = v.z; d[3] = v.w;
        }
        if (tid < EDGES_PER_BLOCK) {
            int e2 = eb + tid;
#pragma unroll
            for (int j = 0; j < N_PAIR; ++j) pfLds[tid][j] = pf[e2 * N_PAIR + j];
            pfLds[tid][N_PAIR] = 1.0f;          // bias slice weight
            a0Lds[tid] = a2p[2 * e2];           // destination atom
        }
    }
    __syncthreads();

    const int h0   = wave * 16;          // this wave's 16-column h block
    const int n    = lane & 15;          // N / M-row index within half-wave
    const int koff = (lane >> 4) * 2;    // K offset per ISA A/B fragment layout
    const int mrow = (lane >> 4) * 8;    // C/D row base per ISA layout

    // ---- preload A fragments: af[t][ks] covers K = ks*4 .. ks*4+3 ----
    v2f af[TILES][16];
#pragma unroll
    for (int t = 0; t < TILES; ++t)
#pragma unroll
        for (int ks = 0; ks < 16; ++ks)
            af[t][ks] = *(const v2f*)&srcLds[t * 16 + n][ks * 4 + koff];

    v8f accO[TILES];
#pragma unroll
    for (int t = 0; t < TILES; ++t) accO[t] = (v8f)0.0f;

    // ---- 17 slices: p = 0..15 weighted by pf, p = 16 is the bias slice ----
    for (int p = 0; p <= N_PAIR; ++p) {
        const float* wp = Wt + (size_t)p * (NH * NH);   // [k][h] layout
        v2f bf[16];
#pragma unroll
        for (int ks = 0; ks < 16; ++ks) {
            int kb   = ks * 4 + koff;
            bf[ks].x = wp[kb * NH + h0 + n];            // coalesced 64B per half-wave
            bf[ks].y = wp[(kb + 1) * NH + h0 + n];
        }
#pragma unroll
        for (int t = 0; t < TILES; ++t) {
            v8f acc = (v8f)0.0f;
#pragma unroll
            for (int ks = 0; ks < 16; ++ks)
                acc = __builtin_amdgcn_wmma_f32_16x16x4_f32(
                    false, af[t][ks], false, bf[ks], (short)0, acc, false, false);
#pragma unroll
            for (int i = 0; i < 8; ++i) {
                float w = pfLds[t * 16 + mrow + i][p];  // LDS broadcast read
                accO[t][i] += w * acc[i];
            }
        }
    }

    // ---- segment-sum scatter: global_atomic_add_f32 ----
#pragma unroll
    for (int t = 0; t < TILES; ++t)
#pragma unroll
        for (int i = 0; i < 8; ++i) {
            int el   = t * 16 + mrow + i;
            int atom = a0Lds[el];
            atomicAdd(&out[(size_t)atom * NH + h0 + n], accO[t][i]);
        }
}

extern "C" void kernel_launch(void* const* d_in, const int* in_sizes, int n_in,
                              void* d_out, int out_size, void* d_ws, size_t ws_size,
                              hipStream_t stream) {
    const float* pf    = (const float*)d_in[0];   // (65536,16)
    const float* atomf = (const float*)d_in[1];   // (8192,64)
    const int*   a2p   = (const int*)d_in[2];     // (65536,2)
    const float* W     = (const float*)d_in[3];   // (16,4096)
    const float* b     = (const float*)d_in[4];   // (4096,)
    float* out = (float*)d_out;                   // (8192,64)
    float* Wt  = (float*)d_ws;                    // 17*64*64 floats = 272 KB

    int prep_n = 17 * NH * NH;
    prep_wt<<<(prep_n + 255) / 256, 256, 0, stream>>>(W, b, Wt);
    zero_out<<<(N_ATOMS * NH + 255) / 256, 256, 0, stream>>>(out, N_ATOMS * NH);
    edge_net<<<N_EDGES / EDGES_PER_BLOCK, 128, 0, stream>>>(pf, atomf, a2p, Wt, out);
}